// GAT_59416577573241
// MI455X (gfx1250) — compile-verified
//
#include <hip/hip_runtime.h>
#include <hip/hip_bf16.h>

// ---------------- problem constants ----------------
#define NN      50000
#define F_IN    128
#define HID     32
#define HEADS   4
#define N_CLS   40
#define EDG     800000LL
#define EI_TOT  (EDG + NN)          // edges + self loops = 850000
#define NEG_SLOPE 0.2f

typedef __attribute__((ext_vector_type(2))) float v2f;
typedef __attribute__((ext_vector_type(8))) float v8f;

// order-preserving float<->uint map for atomicMax on floats
__device__ __forceinline__ unsigned mapf(float f) {
    unsigned u = __float_as_uint(f);
    return (u & 0x80000000u) ? ~u : (u | 0x80000000u);
}
__device__ __forceinline__ float unmapf(unsigned u) {
    unsigned v = (u & 0x80000000u) ? (u ^ 0x80000000u) : ~u;
    return __uint_as_float(v);
}
__device__ __forceinline__ float lrelu(float v) {
    return v > 0.0f ? v : NEG_SLOPE * v;
}
__device__ __forceinline__ void edge_sd(const long long* __restrict__ ei,
                                        long long e, int& s, int& d) {
    if (e < EDG) { s = (int)ei[e]; d = (int)ei[EDG + e]; }
    else         { s = d = (int)(e - EDG); }
}

// ---------------- GEMM1: H[N,128] = X[N,128] @ W1[128,128] (FP32 WMMA) -------
__global__ __launch_bounds__(256)
void gat_gemm1(const float* __restrict__ X, const float* __restrict__ W,
               float* __restrict__ H) {
    const int lane = threadIdx.x;            // 0..31
    const int m0   = blockIdx.x * 16;        // row tile
    const int n0   = threadIdx.y * 16;       // col tile (8 waves cover 128 cols)
    const int row  = lane & 15;
    const int kofs = (lane >> 4) << 1;       // 0 or 2 (A/B k-striping per ISA)

    v8f acc = {0.f,0.f,0.f,0.f,0.f,0.f,0.f,0.f};
    const float* xrow = X + (size_t)(m0 + row) * F_IN;

    #pragma unroll 4
    for (int kb = 0; kb < F_IN / 4; ++kb) {
        const int k = kb * 4 + kofs;
        v2f a, b;
        a.x = xrow[k];
        a.y = xrow[k + 1];
        b.x = W[(size_t)k       * 128 + n0 + row];
        b.y = W[(size_t)(k + 1) * 128 + n0 + row];
        acc = __builtin_amdgcn_wmma_f32_16x16x4_f32(false, a, false, b,
                                                    (short)0, acc, false, false);
    }
    // D layout: VGPR r, lanes 0-15 -> M=r, lanes 16-31 -> M=r+8; N = lane&15
    const int mbase = m0 + ((lane >> 4) << 3);
    float* hp = H + (size_t)mbase * 128 + n0 + row;
    #pragma unroll
    for (int r = 0; r < 8; ++r) hp[(size_t)r * 128] = acc[r];
}

// ---------------- GEMM2: G[N,40] = H2[N,128] @ W2[128,40] (FP32 WMMA) --------
__global__ __launch_bounds__(96)
void gat_gemm2(const float* __restrict__ Hin, const float* __restrict__ W,
               float* __restrict__ G) {
    const int lane = threadIdx.x;
    const int m0   = blockIdx.x * 16;
    const int n0   = threadIdx.y * 16;       // 3 waves cover 40 (padded to 48)
    const int row  = lane & 15;
    const int kofs = (lane >> 4) << 1;
    const int col  = n0 + row;
    const bool okc = (col < N_CLS);

    v8f acc = {0.f,0.f,0.f,0.f,0.f,0.f,0.f,0.f};
    const float* xrow = Hin + (size_t)(m0 + row) * F_IN;

    #pragma unroll 4
    for (int kb = 0; kb < F_IN / 4; ++kb) {
        const int k = kb * 4 + kofs;
        v2f a, b;
        a.x = xrow[k];
        a.y = xrow[k + 1];
        b.x = okc ? W[(size_t)k       * N_CLS + col] : 0.f;   // zero-pad cols
        b.y = okc ? W[(size_t)(k + 1) * N_CLS + col] : 0.f;
        acc = __builtin_amdgcn_wmma_f32_16x16x4_f32(false, a, false, b,
                                                    (short)0, acc, false, false);
    }
    if (okc) {
        const int mbase = m0 + ((lane >> 4) << 3);
        #pragma unroll
        for (int r = 0; r < 8; ++r)
            G[(size_t)(mbase + r) * N_CLS + col] = acc[r];
    }
}

// ------------- per-node attention logits, layer 1 ([N,4]) --------------------
__global__ void gat_attn1(const float* __restrict__ H,
                          const float* __restrict__ att_s,
                          const float* __restrict__ att_d,
                          float* __restrict__ as1, float* __restrict__ ad1) {
    long long t = (long long)blockIdx.x * blockDim.x + threadIdx.x;
    if (t >= (long long)NN * HEADS) return;
    const int h = (int)(t & 3);
    const long long n = t >> 2;
    const float* hp = H + (size_t)n * F_IN + h * HID;
    const float* ws = att_s + h * HID;
    const float* wd = att_d + h * HID;
    float ss = 0.f, sd = 0.f;
    #pragma unroll
    for (int c = 0; c < HID; ++c) { ss += hp[c] * ws[c]; sd += hp[c] * wd[c]; }
    as1[t] = ss; ad1[t] = sd;
}

// ------------- per-node attention logits, layer 2 ([N]) ----------------------
__global__ void gat_attn2(const float* __restrict__ G,
                          const float* __restrict__ att_s,
                          const float* __restrict__ att_d,
                          float* __restrict__ as2, float* __restrict__ ad2) {
    long long n = (long long)blockIdx.x * blockDim.x + threadIdx.x;
    if (n >= NN) return;
    const float* gp = G + (size_t)n * N_CLS;
    float ss = 0.f, sd = 0.f;
    #pragma unroll
    for (int c = 0; c < N_CLS; ++c) { ss += gp[c] * att_s[c]; sd += gp[c] * att_d[c]; }
    as2[n] = ss; ad2[n] = sd;
}

// ------------- edge passes, layer 1 ------------------------------------------
__global__ void gat_edge_max1(const long long* __restrict__ ei,
                              const float* __restrict__ as1,
                              const float* __restrict__ ad1,
                              unsigned* __restrict__ m1) {
    long long t = (long long)blockIdx.x * blockDim.x + threadIdx.x;
    if (t >= EI_TOT * HEADS) return;
    const int h = (int)(t & 3);
    int s, d; edge_sd(ei, t >> 2, s, d);
    float v = lrelu(as1[s * HEADS + h] + ad1[d * HEADS + h]);
    atomicMax(&m1[d * HEADS + h], mapf(v));
}

__global__ void gat_edge_sum1(const long long* __restrict__ ei,
                              const float* __restrict__ as1,
                              const float* __restrict__ ad1,
                              const unsigned* __restrict__ m1,
                              float* __restrict__ z1) {
    long long t = (long long)blockIdx.x * blockDim.x + threadIdx.x;
    if (t >= EI_TOT * HEADS) return;
    const int h = (int)(t & 3);
    int s, d; edge_sd(ei, t >> 2, s, d);
    float v = lrelu(as1[s * HEADS + h] + ad1[d * HEADS + h]);
    atomicAdd(&z1[d * HEADS + h], expf(v - unmapf(m1[d * HEADS + h])));
}

__global__ void gat_edge_aggr1(const long long* __restrict__ ei,
                               const float* __restrict__ as1,
                               const float* __restrict__ ad1,
                               const unsigned* __restrict__ m1,
                               const float* __restrict__ z1,
                               const float* __restrict__ H,
                               float* __restrict__ out1) {
    long long t = (long long)blockIdx.x * blockDim.x + threadIdx.x;
    if (t >= EI_TOT * F_IN) return;
    const int j = (int)(t & (F_IN - 1));     // channel within [H*C]=128
    const int h = j >> 5;                    // head
    int s, d; edge_sd(ei, t >> 7, s, d);
    float v = lrelu(as1[s * HEADS + h] + ad1[d * HEADS + h]);
    float alpha = expf(v - unmapf(m1[d * HEADS + h])) / z1[d * HEADS + h];
    atomicAdd(&out1[(size_t)d * F_IN + j], H[(size_t)s * F_IN + j] * alpha);
}

// ------------- ELU(+b1) in place ---------------------------------------------
__global__ void gat_elu_bias(float* __restrict__ out1, const float* __restrict__ b1) {
    long long t = (long long)blockIdx.x * blockDim.x + threadIdx.x;
    if (t >= (long long)NN * F_IN) return;
    float v = out1[t] + b1[(int)(t & (F_IN - 1))];
    out1[t] = v > 0.f ? v : (expf(v) - 1.f);
}

// ------------- fill output with b2 (atomics accumulate on top) ---------------
__global__ void gat_fill_b2(float* __restrict__ out, const float* __restrict__ b2) {
    long long t = (long long)blockIdx.x * blockDim.x + threadIdx.x;
    if (t >= (long long)NN * N_CLS) return;
    out[t] = b2[(int)(t % N_CLS)];
}

// ------------- edge passes, layer 2 (single head) ----------------------------
__global__ void gat_edge_max2(const long long* __restrict__ ei,
                              const float* __restrict__ as2,
                              const float* __restrict__ ad2,
                              unsigned* __restrict__ m2) {
    long long e = (long long)blockIdx.x * blockDim.x + threadIdx.x;
    if (e >= EI_TOT) return;
    int s, d; edge_sd(ei, e, s, d);
    atomicMax(&m2[d], mapf(lrelu(as2[s] + ad2[d])));
}

__global__ void gat_edge_sum2(const long long* __restrict__ ei,
                              const float* __restrict__ as2,
                              const float* __restrict__ ad2,
                              const unsigned* __restrict__ m2,
                              float* __restrict__ z2) {
    long long e = (long long)blockIdx.x * blockDim.x + threadIdx.x;
    if (e >= EI_TOT) return;
    int s, d; edge_sd(ei, e, s, d);
    float v = lrelu(as2[s] + ad2[d]);
    atomicAdd(&z2[d], expf(v - unmapf(m2[d])));
}

__global__ void gat_edge_aggr2(const long long* __restrict__ ei,
                               const float* __restrict__ as2,
                               const float* __restrict__ ad2,
                               const unsigned* __restrict__ m2,
                               const float* __restrict__ z2,
                               const float* __restrict__ G,
                               float* __restrict__ out) {
    long long t = (long long)blockIdx.x * blockDim.x + threadIdx.x;
    if (t >= EI_TOT * N_CLS) return;
    const int c = (int)(t % N_CLS);
    int s, d; edge_sd(ei, t / N_CLS, s, d);
    float v = lrelu(as2[s] + ad2[d]);
    float alpha = expf(v - unmapf(m2[d])) / z2[d];
    atomicAdd(&out[(size_t)d * N_CLS + c], G[(size_t)s * N_CLS + c] * alpha);
}

// ---------------- host launcher ----------------------------------------------
static inline int nblk(long long n, int b) { return (int)((n + b - 1) / b); }

extern "C" void kernel_launch(void* const* d_in, const int* in_sizes, int n_in,
                              void* d_out, int out_size, void* d_ws, size_t ws_size,
                              hipStream_t stream) {
    (void)in_sizes; (void)n_in; (void)out_size; (void)ws_size;

    const float* x        = (const float*)d_in[0];
    const float* W1       = (const float*)d_in[1];
    const float* att_src1 = (const float*)d_in[2];
    const float* att_dst1 = (const float*)d_in[3];
    const float* b1       = (const float*)d_in[4];
    const float* W2       = (const float*)d_in[5];
    const float* att_src2 = (const float*)d_in[6];
    const float* att_dst2 = (const float*)d_in[7];
    const float* b2       = (const float*)d_in[8];
    const long long* ei   = (const long long*)d_in[9];
    float* out            = (float*)d_out;

    // workspace layout (float offsets)
    float* ws = (float*)d_ws;
    float*    H1  = ws;                         // N*128
    float*    O1  = ws + 6400000;               // N*128 (atomic accum / ELU'd)
    float*    G2  = ws + 12800000;              // N*40
    float*    AS1 = ws + 14800000;              // N*4
    float*    AD1 = ws + 15000000;              // N*4
    unsigned* M1  = (unsigned*)(ws + 15200000); // N*4
    float*    Z1  = ws + 15400000;              // N*4
    float*    AS2 = ws + 15600000;              // N
    float*    AD2 = ws + 15650000;              // N
    unsigned* M2  = (unsigned*)(ws + 15700000); // N
    float*    Z2  = ws + 15750000;              // N   (end: 63.2 MB)

    // zero accumulators (M* zero == mapped -NaN, below every real value)
    hipMemsetAsync(O1, 0, (size_t)NN * F_IN * sizeof(float), stream);
    hipMemsetAsync(M1, 0, (size_t)NN * HEADS * 2 * sizeof(float), stream); // M1+Z1
    hipMemsetAsync(M2, 0, (size_t)NN * 2 * sizeof(float), stream);         // M2+Z2

    // ---- layer 1 ----
    gat_gemm1<<<NN / 16, dim3(32, 8), 0, stream>>>(x, W1, H1);
    gat_attn1<<<nblk((long long)NN * HEADS, 256), 256, 0, stream>>>(H1, att_src1, att_dst1, AS1, AD1);
    gat_edge_max1<<<nblk(EI_TOT * HEADS, 256), 256, 0, stream>>>(ei, AS1, AD1, M1);
    gat_edge_sum1<<<nblk(EI_TOT * HEADS, 256), 256, 0, stream>>>(ei, AS1, AD1, M1, Z1);
    gat_edge_aggr1<<<nblk(EI_TOT * F_IN, 256), 256, 0, stream>>>(ei, AS1, AD1, M1, Z1, H1, O1);
    gat_elu_bias<<<nblk((long long)NN * F_IN, 256), 256, 0, stream>>>(O1, b1);

    // ---- layer 2 ----
    gat_gemm2<<<NN / 16, dim3(32, 3), 0, stream>>>(O1, W2, G2);
    gat_attn2<<<nblk(NN, 256), 256, 0, stream>>>(G2, att_src2, att_dst2, AS2, AD2);
    gat_fill_b2<<<nblk((long long)NN * N_CLS, 256), 256, 0, stream>>>(out, b2);
    gat_edge_max2<<<nblk(EI_TOT, 256), 256, 0, stream>>>(ei, AS2, AD2, M2);
    gat_edge_sum2<<<nblk(EI_TOT, 256), 256, 0, stream>>>(ei, AS2, AD2, M2, Z2);
    gat_edge_aggr2<<<nblk(EI_TOT * N_CLS, 256), 256, 0, stream>>>(ei, AS2, AD2, M2, Z2, G2, out);
}